// GAT_15504831938795
// MI455X (gfx1250) — compile-verified
//
#include <hip/hip_runtime.h>

#define HEADS 8
#define HIDC  8
#define DOUT  64   // HEADS * HIDC
#define NSLOPE 0.2f
#define KC    32   // K-chunk staged in LDS
#define KCP   36   // padded LDS row stride (dwords) -> conflict-free stride-16 access

typedef __attribute__((ext_vector_type(2))) float v2f;
typedef __attribute__((ext_vector_type(8))) float v8f;

// ---------------- helpers ----------------

// Monotonic float -> uint mapping so atomicMax(uint) == float max.
__device__ __forceinline__ unsigned f2ord(float f) {
  unsigned u = __float_as_uint(f);
  return (u & 0x80000000u) ? ~u : (u | 0x80000000u);
}
__device__ __forceinline__ float ord2f(unsigned u) {
  unsigned v = (u & 0x80000000u) ? (u & 0x7FFFFFFFu) : ~u;
  return __uint_as_float(v);
}

__global__ void fill_f32(float* __restrict__ p, float v, long long n) {
  long long i = (long long)blockIdx.x * blockDim.x + threadIdx.x;
  if (i < n) p[i] = v;
}
__global__ void fill_u32(unsigned* __restrict__ p, unsigned v, long long n) {
  long long i = (long long)blockIdx.x * blockDim.x + threadIdx.x;
  if (i < n) p[i] = v;
}

// ---------------- dual projection GEMM via WMMA f32 16x16x4 ----------------
// XL = Hin @ Wl^T + bl ; XR = Hin @ Wr^T + br.  Hin: [N, din] row-major,
// W*: [64, din] row-major.  8 waves/block, one 16-row tile per wave.
// Weight K-chunks are staged in LDS; B fragments come from ds_load_b64.
// A (16x4, f32): lane<16 -> row=lane, K={0,1}; lane>=16 -> row=lane-16, K={2,3}
// B (4x16, f32): lane<16 -> col=lane, K={0,1}; lane>=16 -> col=lane-16, K={2,3}
// C/D (16x16 f32): vgpr v, lane -> M = v + (lane>>4)*8, Ncol = lane&15
__global__ void gemm_dual(const float* __restrict__ Hin, int din,
                          const float* __restrict__ Wl, const float* __restrict__ bl,
                          const float* __restrict__ Wr, const float* __restrict__ br,
                          float* __restrict__ XL, float* __restrict__ XR, int n) {
  __shared__ float sW[2][64 * KCP];   // 2 x 9 KB

  int wave = threadIdx.x >> 5;
  int lane = threadIdx.x & 31;
  int row0 = (blockIdx.x * (blockDim.x >> 5) + wave) * 16;
  bool active = row0 < n;             // wave-uniform: EXEC all-1 inside WMMA region

  int r  = lane & 15;
  int kb = (lane >> 4) * 2;

  v8f accL[4] = {};
  v8f accR[4] = {};

  const float* hrow = Hin + (size_t)((active ? row0 : 0) + r) * din + kb;

  for (int kc = 0; kc < din; kc += KC) {
    // cooperative stage: sW[m][oc*KCP + k] = W[oc*din + kc + k], float4 granules
    for (int i = threadIdx.x; i < 64 * (KC / 4); i += blockDim.x) {
      int oc = i / (KC / 4);
      int k  = (i % (KC / 4)) * 4;
      *(float4*)&sW[0][oc * KCP + k] = *(const float4*)(Wl + (size_t)oc * din + kc + k);
      *(float4*)&sW[1][oc * KCP + k] = *(const float4*)(Wr + (size_t)oc * din + kc + k);
    }
    __syncthreads();

    if (active) {
      if (kc + KC < din) __builtin_prefetch(hrow + kc + KC, 0, 1);
#pragma unroll
      for (int k0 = 0; k0 < KC; k0 += 4) {
        v2f a;
        a[0] = hrow[kc + k0 + 0];
        a[1] = hrow[kc + k0 + 1];
#pragma unroll
        for (int t = 0; t < 4; ++t) {
          int oc = t * 16 + r;
          const float* wl = &sW[0][oc * KCP + k0 + kb];
          v2f bvl; bvl[0] = wl[0]; bvl[1] = wl[1];
          accL[t] = __builtin_amdgcn_wmma_f32_16x16x4_f32(
              false, a, false, bvl, (short)0, accL[t], false, false);
          const float* wr = &sW[1][oc * KCP + k0 + kb];
          v2f bvr; bvr[0] = wr[0]; bvr[1] = wr[1];
          accR[t] = __builtin_amdgcn_wmma_f32_16x16x4_f32(
              false, a, false, bvr, (short)0, accR[t], false, false);
        }
      }
    }
    __syncthreads();
  }

  if (!active) return;

  int mbase = (lane >> 4) * 8;
#pragma unroll
  for (int t = 0; t < 4; ++t) {
    int col = t * 16 + (lane & 15);
    float addl = bl[col];
    float addr_ = br[col];
#pragma unroll
    for (int v = 0; v < 8; ++v) {
      size_t o = (size_t)(row0 + mbase + v) * DOUT + col;
      XL[o] = accL[t][v] + addl;
      XR[o] = accR[t][v] + addr_;
    }
  }
}

// ---------------- edge passes (1 thread per (edge, head)) ----------------
// Edges e < E come from edge_index; e >= E are self loops (src=dst=e-E).

__global__ void edge_score(const int* __restrict__ src, const int* __restrict__ dst,
                           const float* __restrict__ XL, const float* __restrict__ XR,
                           const float* __restrict__ att,
                           float* __restrict__ SC, unsigned* __restrict__ M,
                           int E, long long Etot) {
  long long t = (long long)blockIdx.x * blockDim.x + threadIdx.x;
  if (t >= Etot * HEADS) return;
  long long e = t >> 3;
  int hh = (int)(t & 7);
  int s = (e < E) ? src[e] : (int)(e - E);
  int d = (e < E) ? dst[e] : (int)(e - E);
  const float* pl = XL + (size_t)s * DOUT + hh * HIDC;
  const float* pr = XR + (size_t)d * DOUT + hh * HIDC;
  const float* pa = att + hh * HIDC;
  float4 l0 = *(const float4*)(pl);     float4 l1 = *(const float4*)(pl + 4);
  float4 r0 = *(const float4*)(pr);     float4 r1 = *(const float4*)(pr + 4);
  float4 a0 = *(const float4*)(pa);     float4 a1 = *(const float4*)(pa + 4);
  float lv[8] = {l0.x, l0.y, l0.z, l0.w, l1.x, l1.y, l1.z, l1.w};
  float rv[8] = {r0.x, r0.y, r0.z, r0.w, r1.x, r1.y, r1.z, r1.w};
  float av[8] = {a0.x, a0.y, a0.z, a0.w, a1.x, a1.y, a1.z, a1.w};
  float acc = 0.f;
#pragma unroll
  for (int c = 0; c < HIDC; ++c) {
    float v = lv[c] + rv[c];
    v = (v > 0.f) ? v : NSLOPE * v;
    acc += v * av[c];
  }
  SC[t] = acc;
  atomicMax(&M[(size_t)d * HEADS + hh], f2ord(acc));
}

__global__ void edge_exp(const int* __restrict__ src, const int* __restrict__ dst,
                         float* __restrict__ SC, const unsigned* __restrict__ M,
                         float* __restrict__ DEN, int E, long long Etot) {
  long long t = (long long)blockIdx.x * blockDim.x + threadIdx.x;
  if (t >= Etot * HEADS) return;
  long long e = t >> 3;
  int hh = (int)(t & 7);
  int d = (e < E) ? dst[e] : (int)(e - E);
  float m = ord2f(M[(size_t)d * HEADS + hh]);
  float a = __expf(SC[t] - m);
  SC[t] = a;
  atomicAdd(&DEN[(size_t)d * HEADS + hh], a);
}

__global__ void edge_scatter(const int* __restrict__ src, const int* __restrict__ dst,
                             const float* __restrict__ SC, const float* __restrict__ DEN,
                             const float* __restrict__ XL, float* __restrict__ ACC,
                             int E, long long Etot) {
  long long t = (long long)blockIdx.x * blockDim.x + threadIdx.x;
  if (t >= Etot * HEADS) return;
  long long e = t >> 3;
  int hh = (int)(t & 7);
  int s = (e < E) ? src[e] : (int)(e - E);
  int d = (e < E) ? dst[e] : (int)(e - E);
  float alpha = SC[t] / DEN[(size_t)d * HEADS + hh];
  const float* pl = XL + (size_t)s * DOUT + hh * HIDC;
  float4 l0 = *(const float4*)(pl);
  float4 l1 = *(const float4*)(pl + 4);
  float lv[8] = {l0.x, l0.y, l0.z, l0.w, l1.x, l1.y, l1.z, l1.w};
  float* po = ACC + (size_t)d * DOUT + hh * HIDC;
#pragma unroll
  for (int c = 0; c < HIDC; ++c) atomicAdd(&po[c], alpha * lv[c]);
}

// ---------------- per-node finalize ----------------

__device__ __forceinline__ float elu(float x) {
  return (x > 0.f) ? x : (__expf(x) - 1.f);
}

// concat layers: out[n, col] = elu(ACC[n, col] + bias[col])    (col in [0,64))
__global__ void finalize_concat(const float* __restrict__ ACC, const float* __restrict__ bias,
                                float* __restrict__ Hout, int n) {
  long long i = (long long)blockIdx.x * blockDim.x + threadIdx.x;
  if (i >= (long long)n * DOUT) return;
  int col = (int)(i & (DOUT - 1));
  Hout[i] = elu(ACC[i] + bias[col]);
}

// last layer: mean over heads then bias, elu -> [N, 8]
__global__ void finalize_mean(const float* __restrict__ ACC, const float* __restrict__ bias,
                              float* __restrict__ H8, int n) {
  long long i = (long long)blockIdx.x * blockDim.x + threadIdx.x;
  if (i >= (long long)n * HIDC) return;
  int node = (int)(i >> 3);
  int c = (int)(i & 7);
  const float* p = ACC + (size_t)node * DOUT + c;
  float v = 0.f;
#pragma unroll
  for (int h = 0; h < HEADS; ++h) v += p[h * HIDC];
  H8[i] = elu(v * (1.f / HEADS) + bias[c]);
}

// ---------------- readout + head ----------------

__global__ void readout(const float* __restrict__ H8, const int* __restrict__ batch,
                        float* __restrict__ GB, int n) {
  long long i = (long long)blockIdx.x * blockDim.x + threadIdx.x;
  if (i >= (long long)n * HIDC) return;
  int node = (int)(i >> 3);
  int c = (int)(i & 7);
  atomicAdd(&GB[(size_t)batch[node] * HIDC + c], H8[i]);
}

__global__ void head_kernel(const float* __restrict__ GB, const float* __restrict__ hw,
                            const float* __restrict__ hb, float* __restrict__ out, int g) {
  int i = blockIdx.x * blockDim.x + threadIdx.x;
  if (i >= g) return;
  float v = 0.f;
#pragma unroll
  for (int c = 0; c < HIDC; ++c) v += GB[(size_t)i * HIDC + c] * hw[c];
  out[i] = v + hb[0];
}

// ---------------- launch ----------------

static inline int nblk(long long n, int b) { return (int)((n + b - 1) / b); }

extern "C" void kernel_launch(void* const* d_in, const int* in_sizes, int n_in,
                              void* d_out, int out_size, void* d_ws, size_t ws_size,
                              hipStream_t stream) {
  (void)n_in; (void)ws_size;
  const float* x     = (const float*)d_in[0];
  const int*   ei    = (const int*)d_in[1];
  const int*   batch = (const int*)d_in[2];

  const int N  = in_sizes[2];
  const int E  = in_sizes[1] / 2;
  const int IN = in_sizes[0] / N;
  const int G  = out_size;
  const long long Etot = (long long)E + N;

  const int* src = ei;
  const int* dst = ei + E;

  // workspace carve-out
  float* ws = (float*)d_ws;
  const size_t nf = (size_t)N * DOUT;
  float*    XL  = ws;
  float*    XR  = XL + nf;
  float*    ACC = XR + nf;
  float*    H   = ACC + nf;
  float*    SC  = H + nf;
  unsigned* M   = (unsigned*)(SC + (size_t)Etot * HEADS);
  float*    DEN = (float*)(M + (size_t)N * HEADS);
  float*    GB  = DEN + (size_t)N * HEADS;

  const float* hin = x;
  int din = IN;

  const long long n8 = (long long)N * HEADS;
  const long long t8 = Etot * HEADS;
  const int eb = nblk(t8, 256);

  for (int L = 0; L < 4; ++L) {
    const float* Wl   = (const float*)d_in[3 + 6 * L + 0];
    const float* bl   = (const float*)d_in[3 + 6 * L + 1];
    const float* Wr   = (const float*)d_in[3 + 6 * L + 2];
    const float* br   = (const float*)d_in[3 + 6 * L + 3];
    const float* att  = (const float*)d_in[3 + 6 * L + 4];
    const float* bias = (const float*)d_in[3 + 6 * L + 5];

    int tiles = (N + 15) / 16;
    gemm_dual<<<nblk(tiles, 8), 256, 0, stream>>>(hin, din, Wl, bl, Wr, br, XL, XR, N);

    fill_u32<<<nblk(n8, 256), 256, 0, stream>>>(M, 0u, n8);
    fill_f32<<<nblk(n8, 256), 256, 0, stream>>>(DEN, 0.f, n8);
    fill_f32<<<nblk((long long)nf, 256), 256, 0, stream>>>(ACC, 0.f, (long long)nf);

    edge_score  <<<eb, 256, 0, stream>>>(src, dst, XL, XR, att, SC, M, E, Etot);
    edge_exp    <<<eb, 256, 0, stream>>>(src, dst, SC, M, DEN, E, Etot);
    edge_scatter<<<eb, 256, 0, stream>>>(src, dst, SC, DEN, XL, ACC, E, Etot);

    if (L < 3) {
      finalize_concat<<<nblk((long long)nf, 256), 256, 0, stream>>>(ACC, bias, H, N);
      hin = H;
      din = DOUT;
    } else {
      // reuse XL as the final [N,8] feature buffer (XL no longer needed)
      finalize_mean<<<nblk((long long)N * HIDC, 256), 256, 0, stream>>>(ACC, bias, XL, N);
    }
  }

  const float* head_w = (const float*)d_in[3 + 6 * 4 + 0];
  const float* head_b = (const float*)d_in[3 + 6 * 4 + 1];

  fill_f32<<<nblk((long long)G * HIDC, 256), 256, 0, stream>>>(GB, 0.f, (long long)G * HIDC);
  readout<<<nblk((long long)N * HIDC, 256), 256, 0, stream>>>(XL, batch, GB, N);
  head_kernel<<<nblk(G, 128), 128, 0, stream>>>(GB, head_w, head_b, (float*)d_out, G);
}